// SAGE_CONCAT_15676630630848
// MI455X (gfx1250) — compile-verified
//
#include <hip/hip_runtime.h>

typedef float v2f __attribute__((ext_vector_type(2)));
typedef float v4f __attribute__((ext_vector_type(4)));
typedef float v8f __attribute__((ext_vector_type(8)));
typedef int   v4i __attribute__((ext_vector_type(4)));

#define Bc 2
#define Nc 100000
#define Kc 16
#define Fc 128
#define Cc 10

// -------- Kernel 1: gather-mean (mp), L2-resident gather, NT streaming store --------
// One wave (32 lanes) per node; lane owns float4 -> 32*16B = 512B = one row of F=128.
__global__ __launch_bounds__(256) void gather_mean_kernel(
    const float* __restrict__ x, const int* __restrict__ idx, float* __restrict__ mp)
{
    const int lane = threadIdx.x & 31;
    const int wave = threadIdx.x >> 5;
    const int node = blockIdx.x * 8 + wave;          // 8 waves per block
    if (node >= Bc * Nc) return;
    const int b = (node >= Nc) ? 1 : 0;

    const float* xb = x + (size_t)b * Nc * Fc;
    const v4i* nb  = reinterpret_cast<const v4i*>(idx + (size_t)node * Kc);

    v4f acc = (v4f)0.0f;
#pragma unroll
    for (int q = 0; q < 4; ++q) {
        v4i j4 = nb[q];                              // 16B index load
#pragma unroll
        for (int t = 0; t < 4; ++t) {
            const v4f* row = reinterpret_cast<const v4f*>(xb + (size_t)j4[t] * Fc);
            acc += row[lane];                        // global_load_b128, L2 hit
        }
    }
    acc *= 0.0625f;                                  // mean over K=16
    v4f* out = reinterpret_cast<v4f*>(mp + (size_t)node * Fc);
    __builtin_nontemporal_store(acc, out + lane);    // NT b128: don't pollute L2
}

// -------- Kernel 2: head MLP via V_WMMA_F32_16X16X4_F32 (single wave) --------
__global__ __launch_bounds__(32) void head_kernel(
    const float* __restrict__ x,  const int* __restrict__ idx,
    const float* __restrict__ W1, const float* __restrict__ b1,
    const float* __restrict__ W2, const float* __restrict__ b2,
    const float* __restrict__ Wo, const float* __restrict__ bo,
    float* __restrict__ probs)
{
    __shared__ float Amat[2][16][128];   // [0]=old_vec (mean), [1]=new_vec; rows 2..15 = 0
    __shared__ float h[2][128];          // concat([relu(old@W1+b1), relu(new@W2+b2)])
    __shared__ float lg[2][16];
    const int lane = threadIdx.x;        // 0..31, one wave

    // Zero the padded A tiles.
    for (int s = 0; s < 2; ++s)
        for (int row = 0; row < 16; ++row)
            for (int c = lane; c < 128; c += 32)
                Amat[s][row][c] = 0.0f;

    // Rows 0..1: recompute old_vec (mean of node-0 neighbors) and copy new_vec.
    for (int b = 0; b < 2; ++b) {
        const float* xb = x + (size_t)b * Nc * Fc;
        const int*   nb = idx + (size_t)b * Nc * Kc;   // neighbors of node (b, 0)
        v4f acc = (v4f)0.0f;
        for (int k = 0; k < Kc; ++k) {
            const v4f* row = reinterpret_cast<const v4f*>(xb + (size_t)nb[k] * Fc);
            acc += row[lane];
        }
        acc *= 0.0625f;
        v4f nv = reinterpret_cast<const v4f*>(xb)[lane];   // x_feats[b, 0, :]
#pragma unroll
        for (int t = 0; t < 4; ++t) {
            Amat[0][b][lane * 4 + t] = acc[t];
            Amat[1][b][lane * 4 + t] = nv[t];
        }
    }
    __syncthreads();

    // GEMM: D(16x16) += A(16x4) x B(4x16), K=128 in steps of 4, N=64 in 4 tiles.
    // A frag layout: lane (half,r): {A[r][k0+2*half], A[r][k0+2*half+1]}
    // B frag layout: lane (half,r): {W[k0+2*half][n0+r], W[k0+2*half+1][n0+r]}
    const int half = lane >> 4;
    const int r    = lane & 15;
    for (int s = 0; s < 2; ++s) {
        const float* W    = s ? W2 : W1;
        const float* bias = s ? b2 : b1;
        for (int t = 0; t < 4; ++t) {
            v8f acc = (v8f)0.0f;
            for (int k0 = 0; k0 < 128; k0 += 4) {
                const int k = k0 + 2 * half;
                v2f av; av.x = Amat[s][r][k];
                        av.y = Amat[s][r][k + 1];
                v2f bv; bv.x = W[(size_t)k       * 64 + t * 16 + r];
                        bv.y = W[(size_t)(k + 1) * 64 + t * 16 + r];
                acc = __builtin_amdgcn_wmma_f32_16x16x4_f32(
                    /*neg_a=*/false, av, /*neg_b=*/false, bv,
                    /*c_mod=*/(short)0, acc, /*reuse_a=*/false, /*reuse_b=*/false);
            }
            // D rows M=0,1 live in acc[0], acc[1] of lanes 0..15.
            if (half == 0) {
                float va = acc[0] + bias[t * 16 + r];
                float vb = acc[1] + bias[t * 16 + r];
                h[0][s * 64 + t * 16 + r] = va > 0.0f ? va : 0.0f;
                h[1][s * 64 + t * 16 + r] = vb > 0.0f ? vb : 0.0f;
            }
        }
    }
    __syncthreads();

    // logits = h @ Wo + bo  (2x128 @ 128x10): 20 lanes, one dot each.
    if (lane < 2 * Cc) {
        const int b = lane / Cc, c = lane % Cc;
        float s = bo[c];
        for (int j = 0; j < 128; ++j) s += h[b][j] * Wo[j * Cc + c];
        lg[b][c] = s;
    }
    __syncthreads();

    // softmax per batch row.
    if (lane < Bc) {
        float m = lg[lane][0];
        for (int c = 1; c < Cc; ++c) m = fmaxf(m, lg[lane][c]);
        float e[Cc], sum = 0.0f;
        for (int c = 0; c < Cc; ++c) { e[c] = expf(lg[lane][c] - m); sum += e[c]; }
        const float inv = 1.0f / sum;
        for (int c = 0; c < Cc; ++c) probs[lane * Cc + c] = e[c] * inv;
    }
}

extern "C" void kernel_launch(void* const* d_in, const int* in_sizes, int n_in,
                              void* d_out, int out_size, void* d_ws, size_t ws_size,
                              hipStream_t stream) {
    const float* x   = (const float*)d_in[0];
    const int*   idx = (const int*)  d_in[1];
    const float* W1  = (const float*)d_in[2];
    const float* b1  = (const float*)d_in[3];
    const float* W2  = (const float*)d_in[4];
    const float* b2  = (const float*)d_in[5];
    const float* Wo  = (const float*)d_in[6];
    const float* bo  = (const float*)d_in[7];

    float* out   = (float*)d_out;
    float* probs = out;             // first B*C = 20 floats
    float* mp    = out + Bc * Cc;   // then B*N*F floats (byte offset 80, 16B aligned)

    const int totalNodes = Bc * Nc;
    const int blocks = (totalNodes + 7) / 8;   // 8 nodes (waves) per 256-thread block
    gather_mean_kernel<<<blocks, 256, 0, stream>>>(x, idx, mp);
    head_kernel<<<1, 32, 0, stream>>>(x, idx, W1, b1, W2, b2, Wo, bo, probs);
}